// Pooler3d_19739669692439
// MI455X (gfx1250) — compile-verified
//
#include <hip/hip_runtime.h>

typedef __attribute__((ext_vector_type(2))) float v2f;
typedef __attribute__((ext_vector_type(8))) float v8f;

#if defined(__has_builtin)
#if __has_builtin(__builtin_amdgcn_s_wait_asynccnt)
#define HAVE_WAIT_ASYNC 1
#endif
#endif

#define OUT_HW 7
#define ROI_SCALE 0.0625f
#define CCH 256
#define TT 16
#define HH 64
#define WW 64
#define NPLANE (CCH * TT)      // 4096 (c,t) planes per image
#define PLANES 16              // planes handled per block (one WMMA N-tile)
#define WIN 20                 // padded ROI window (rows/cols); worst case 17 used

// Separable ROIAlign3d via fp32 WMMA:
//   out[7,7] = Wy(7 x rows) * patch(rows x cols) * Wx^T(cols x 7)  per (roi, plane)
// Stage 1: tmp[ph][plane][col] = sum_r Wy[ph][r] * patch[plane][r][col]   (WMMA per col)
// Stage 2: out[pw][plane]      = sum_c Wx[pw][c] * tmp[ph][plane][c]      (WMMA per ph)
//
// WMMA f32 16x16x4 operand layouts (per CDNA5 ISA 7.12.2):
//   A (16x4): lane L<16 -> M=L, VGPR{0,1}=K{0,1}; lane L>=16 -> M=L-16, K{2,3}
//   B (4x16): rows striped across lanes; assumed mirrored K split:
//             VGPR0: lanes<16 K=0, lanes>=16 K=2 ; VGPR1: K=1 / K=3  (N = lane&15)
//   C/D (16x16): VGPR v: lanes<16 -> M=v, N=lane ; lanes>=16 -> M=v+8, N=lane-16

__global__ __launch_bounds__(128) void roi_align3d_wmma_kernel(
    const float* __restrict__ x,     // [2, 256, 16, 64, 64]
    const float* __restrict__ rois,  // [R, 5]
    float* __restrict__ out)         // [R, 4096, 7, 7]
{
    __shared__ float sWy[16][WIN];               // row weights (ph x r); rows>=7 zero
    __shared__ float sWx[16][WIN];               // col weights (pw x c)
    __shared__ float sPatch[PLANES][WIN][WIN];   // gathered ROI window per plane
    __shared__ float sTmp[16][PLANES][WIN];      // stage-1 result [ph][plane][col]

    const int pc  = blockIdx.x;      // plane chunk: planes pc*16 .. pc*16+15
    const int roi = blockIdx.y;
    const int tid = threadIdx.x;
    const int lane = tid & 31;
    const int wave = tid >> 5;

    // ---- ROI parameters (computed redundantly by every thread; deterministic) ----
    const float rb = rois[roi * 5 + 0];
    const float sw = rois[roi * 5 + 1] * ROI_SCALE;
    const float sh = rois[roi * 5 + 2] * ROI_SCALE;
    const float ew = rois[roi * 5 + 3] * ROI_SCALE;
    const float eh = rois[roi * 5 + 4] * ROI_SCALE;
    const int b = (int)rb;
    const float roi_w = fmaxf(ew - sw, 1.0f);
    const float roi_h = fmaxf(eh - sh, 1.0f);
    const float bin_h = roi_h / (float)OUT_HW;
    const float bin_w = roi_w / (float)OUT_HW;

    // Window origin: clamped-floor of the first sample; extent: corner of last sample.
    const float y_first = fminf(fmaxf(sh + 0.25f * bin_h, 0.0f), (float)(HH - 1));
    const float x_first = fminf(fmaxf(sw + 0.25f * bin_w, 0.0f), (float)(WW - 1));
    const int r0 = (int)floorf(y_first);
    const int c0 = (int)floorf(x_first);
    const float y_lastf = fminf(fmaxf(sh + 6.75f * bin_h, 0.0f), (float)(HH - 1));
    const float x_lastf = fminf(fmaxf(sw + 6.75f * bin_w, 0.0f), (float)(WW - 1));
    const int y1l = min((int)floorf(y_lastf) + 1, HH - 1);
    const int x1l = min((int)floorf(x_lastf) + 1, WW - 1);
    const int rows_used = min(y1l - r0 + 1, WIN);   // all rows r0..r0+rows_used-1 in [0,63]
    const int cols_used = min(x1l - c0 + 1, WIN);

    // ---- zero weight tables ----
    for (int i = tid; i < 16 * WIN; i += 128) {
        sWy[i / WIN][i % WIN] = 0.0f;
        sWx[i / WIN][i % WIN] = 0.0f;
    }
    __syncthreads();

    // ---- build separable interpolation weights (7 threads per axis) ----
    // factor 0.5 per axis -> 0.25 total = 1/(sr*sr); validity mask folded in.
    if (tid < OUT_HW) {
        const int ph = tid;
        for (int g = 0; g < 2; ++g) {
            const float yy = sh + ((float)ph + 0.25f + 0.5f * (float)g) * bin_h;
            const bool valid = (yy >= -1.0f) && (yy <= (float)HH);
            const float yc = fminf(fmaxf(yy, 0.0f), (float)(HH - 1));
            const int y0 = (int)floorf(yc);
            const int y1 = min(y0 + 1, HH - 1);
            const float ly = yc - (float)y0;
            if (valid) {
                int i0 = min(max(y0 - r0, 0), WIN - 1);
                int i1 = min(max(y1 - r0, 0), WIN - 1);
                sWy[ph][i0] += 0.5f * (1.0f - ly);
                sWy[ph][i1] += 0.5f * ly;
            }
        }
    } else if (tid >= 16 && tid < 16 + OUT_HW) {
        const int pw = tid - 16;
        for (int g = 0; g < 2; ++g) {
            const float xx = sw + ((float)pw + 0.25f + 0.5f * (float)g) * bin_w;
            const bool valid = (xx >= -1.0f) && (xx <= (float)WW);
            const float xc = fminf(fmaxf(xx, 0.0f), (float)(WW - 1));
            const int x0 = (int)floorf(xc);
            const int x1 = min(x0 + 1, WW - 1);
            const float lx = xc - (float)x0;
            if (valid) {
                int i0 = min(max(x0 - c0, 0), WIN - 1);
                int i1 = min(max(x1 - c0, 0), WIN - 1);
                sWx[pw][i0] += 0.5f * (1.0f - lx);
                sWx[pw][i1] += 0.5f * lx;
            }
        }
    }

    // ---- stage the exact ROI window into LDS; zero the padding ----
    // Used region: async global->LDS (ASYNCcnt path, no VGPR round trip), via inline asm:
    //   global_load_async_to_lds_b32 vdst(LDS byte addr), vaddr(64b global addr), off
    // LDS byte address = low 32 bits of the generic pointer (ISA 10.2 aperture rule).
    // Unused region: DS zero-stores (disjoint addresses -> no DS/ASYNC ordering hazard).
    {
        const size_t img_base = ((size_t)b * NPLANE + (size_t)pc * PLANES) * (size_t)(HH * WW);
        for (int idx = tid; idx < PLANES * WIN * WIN; idx += 128) {
            const int plane = idx / (WIN * WIN);
            const int rem = idx - plane * (WIN * WIN);
            const int rr = rem / WIN;
            const int cc = rem - rr * WIN;
            if (rr < rows_used && cc < cols_used) {
                const float* g = x + img_base + (size_t)plane * (HH * WW)
                                 + (size_t)(r0 + rr) * WW + (c0 + cc);
                const unsigned lds_addr = (unsigned)(size_t)&sPatch[plane][rr][cc];
                const unsigned long long gaddr = (unsigned long long)(size_t)g;
                asm volatile("global_load_async_to_lds_b32 %0, %1, off"
                             :
                             : "v"(lds_addr), "v"(gaddr)
                             : "memory");
            } else {
                sPatch[plane][rr][cc] = 0.0f;
            }
        }
    }
#if defined(HAVE_WAIT_ASYNC)
    __builtin_amdgcn_s_wait_asynccnt(0);
#else
    asm volatile("s_wait_asynccnt 0" ::: "memory");
#endif
    __syncthreads();

    const int kmax1 = (rows_used + 3) >> 2;   // K chunks over rows (wave-uniform)
    const int kmax2 = (cols_used + 3) >> 2;   // K chunks over cols

    // ---- Stage 1: tmp[ph][plane][col] = Wy * patch, one WMMA N-tile per column ----
    const int mcol = lane & 15;       // M row (A) / N col (B,D) within half-wave
    const int khalf = lane >> 4;      // selects K pair {0,1} vs {2,3}
    #pragma unroll
    for (int it = 0; it < WIN / 4; ++it) {
        const int col = wave + 4 * it;          // wave-uniform, always < WIN
        v8f acc = {0.f, 0.f, 0.f, 0.f, 0.f, 0.f, 0.f, 0.f};
        for (int kk = 0; kk < kmax1; ++kk) {
            const int k0 = 4 * kk + 2 * khalf;
            v2f a, bb;
            a.x = sWy[mcol][k0];
            a.y = sWy[mcol][k0 + 1];
            bb.x = sPatch[mcol][k0][col];
            bb.y = sPatch[mcol][k0 + 1][col];
            acc = __builtin_amdgcn_wmma_f32_16x16x4_f32(
                false, a, false, bb, (short)0, acc, false, false);
        }
        const int mbase = khalf * 8;
        #pragma unroll
        for (int v = 0; v < 8; ++v)
            sTmp[mbase + v][mcol][col] = acc[v];   // rows ph>=7 are zeros, never read
    }
    __syncthreads();

    // ---- Stage 2: out[pw][plane] = Wx * tmp[ph], one WMMA N-tile per ph ----
    #pragma unroll
    for (int it = 0; it < 2; ++it) {
        const int ph = wave + 4 * it;           // wave-uniform guard; EXEC full inside
        if (ph < OUT_HW) {
            v8f acc = {0.f, 0.f, 0.f, 0.f, 0.f, 0.f, 0.f, 0.f};
            for (int kk = 0; kk < kmax2; ++kk) {
                const int k0 = 4 * kk + 2 * khalf;
                v2f a, bb;
                a.x = sWx[mcol][k0];
                a.y = sWx[mcol][k0 + 1];
                bb.x = sTmp[ph][mcol][k0];
                bb.y = sTmp[ph][mcol][k0 + 1];
                acc = __builtin_amdgcn_wmma_f32_16x16x4_f32(
                    false, a, false, bb, (short)0, acc, false, false);
            }
            // D[pw, plane]: pw 0..6 live entirely in lanes 0..15 (M=v)
            if (khalf == 0) {
                const size_t base =
                    (((size_t)roi * NPLANE) + (size_t)pc * PLANES + mcol) * (OUT_HW * OUT_HW)
                    + (size_t)ph * OUT_HW;
                #pragma unroll
                for (int v = 0; v < OUT_HW; ++v)
                    out[base + v] = acc[v];
            }
        }
    }
}

extern "C" void kernel_launch(void* const* d_in, const int* in_sizes, int n_in,
                              void* d_out, int out_size, void* d_ws, size_t ws_size,
                              hipStream_t stream) {
    (void)n_in; (void)out_size; (void)d_ws; (void)ws_size;
    const float* x = (const float*)d_in[0];
    const float* rois = (const float*)d_in[1];
    float* out = (float*)d_out;
    const int R = in_sizes[1] / 5;

    dim3 grid(NPLANE / PLANES, R);   // (256, 64)
    dim3 block(128);                 // 4 waves of 32
    roi_align3d_wmma_kernel<<<grid, block, 0, stream>>>(x, rois, out);
}